// RSTArgumentMiner_36799279792564
// MI455X (gfx1250) — compile-verified
//
#include <hip/hip_runtime.h>

// ---------------------------------------------------------------------------
// RGCN 2-layer forward for MI455X (gfx1250), fp32 WMMA path, atomic-free
// aggregation via per-relation CSR (built on device once per call).
//
// Math rewrite vs reference:
//   msg_e = (x[src_e] * nw[src_e]) @ W_rel[r]  ==  Y_r[src_e],
//   where Y_r = (diag(nw) X) @ W_rel[r]   (node-level GEMM, not edge-level)
//   out[t] += (1/max(cnt_r[t],1)) * sum_{e->t} Y_r[src_e]   (CSR gather, no atomics)
// ---------------------------------------------------------------------------

typedef __attribute__((ext_vector_type(2))) float v2f;
typedef __attribute__((ext_vector_type(8))) float v8f;

#define NN   50000
#define RR   8
#define EE   100000
#define DIN  768
#define DH   256
#define LNE  1e-5f

// ---------------------------------------------------------------------------
// GEMM: C[M x 256] = (SCALE ? diag(rowscale) : I) * A[M x K] @ B[K x 256] (+bias)
// Block: 256 thr = 8 waves, tile 128(M) x 64(N). Wave: 32x32 via 2x2 WMMA tiles.
// V_WMMA_F32_16X16X4_F32: A 16x4 (v2f/lane), B 4x16 (v2f/lane), C 16x16 (v8f/lane).
// A frag: lane L -> row (L&15), k = 2*(L>>4) + {0,1}   (ISA 7.12.2, 32-bit A 16x4)
// B frag: lane L -> col (L&15), k = 2*(L>>4) + {0,1}   (mirrored)
// C frag: vgpr i, lane L -> row i + 8*(L>>4), col (L&15)
// ---------------------------------------------------------------------------
template<bool SCALE, bool BIAS>
__global__ __launch_bounds__(256)
void rgcn_gemm_wmma_f32(const float* __restrict__ A, const float* __restrict__ B,
                        const float* __restrict__ bias, const float* __restrict__ rowscale,
                        float* __restrict__ C, int M, int K) {
    const int lane  = threadIdx.x & 31;
    const int wave  = threadIdx.x >> 5;
    const int wm    = wave & 3;                 // 4 M-subtiles of 32
    const int wn    = wave >> 2;                // 2 N-subtiles of 32
    const int rowBase = blockIdx.x * 128 + wm * 32;
    const int colBase = blockIdx.y * 64  + wn * 32;
    const int mIdx  = lane & 15;
    const int kHalf = lane >> 4;                // 0 or 1

    const int r0 = rowBase + mIdx;              // rows for M-tile 0
    const int r1 = r0 + 16;                     // rows for M-tile 1
    const int r0c = (r0 < M) ? r0 : 0;          // clamp loads (stores are guarded)
    const int r1c = (r1 < M) ? r1 : 0;
    const float s0 = SCALE ? rowscale[r0c] : 1.0f;
    const float s1 = SCALE ? rowscale[r1c] : 1.0f;

    const float* arow0 = A + (size_t)r0c * K + 2 * kHalf;
    const float* arow1 = A + (size_t)r1c * K + 2 * kHalf;
    const int col0 = colBase + mIdx;            // N-tile 0 column
    const int col1 = col0 + 16;                 // N-tile 1 column
    const float* bp = B + (size_t)(2 * kHalf) * DH;

    v8f c00 = {}, c01 = {}, c10 = {}, c11 = {};

    #pragma unroll 4
    for (int k = 0; k < K; k += 4) {
        v2f a0 = *(const v2f*)(arow0 + k);      // 8B-aligned b64 load
        v2f a1 = *(const v2f*)(arow1 + k);
        if (SCALE) { a0 *= s0; a1 *= s1; }
        const float* bk = bp + (size_t)k * DH;
        v2f b0 = { bk[col0], bk[DH + col0] };
        v2f b1 = { bk[col1], bk[DH + col1] };
        c00 = __builtin_amdgcn_wmma_f32_16x16x4_f32(false, a0, false, b0, (short)0, c00, false, false);
        c01 = __builtin_amdgcn_wmma_f32_16x16x4_f32(false, a0, false, b1, (short)0, c01, false, false);
        c10 = __builtin_amdgcn_wmma_f32_16x16x4_f32(false, a1, false, b0, (short)0, c10, false, false);
        c11 = __builtin_amdgcn_wmma_f32_16x16x4_f32(false, a1, false, b1, (short)0, c11, false, false);
    }

    const float bb0 = BIAS ? bias[col0] : 0.0f;
    const float bb1 = BIAS ? bias[col1] : 0.0f;
    const int rb = rowBase + 8 * kHalf;
    #pragma unroll
    for (int i = 0; i < 8; ++i) {
        const int ra = rb + i;                  // M-tile 0 row
        const int rc = ra + 16;                 // M-tile 1 row
        if (ra < M) {
            C[(size_t)ra * DH + col0] = c00[i] + bb0;
            C[(size_t)ra * DH + col1] = c01[i] + bb1;
        }
        if (rc < M) {
            C[(size_t)rc * DH + col0] = c10[i] + bb0;
            C[(size_t)rc * DH + col1] = c11[i] + bb1;
        }
    }
}

// ---------------------------------------------------------------------------
// Setup helpers: zero-fill, u32 degree histogram, reciprocal of clipped count
// ---------------------------------------------------------------------------
__global__ void rgcn_fill_u32_kernel(unsigned* __restrict__ p, unsigned v, int n) {
    int i = blockIdx.x * blockDim.x + threadIdx.x;
    if (i < n) p[i] = v;
}

__global__ void rgcn_count_kernel(const int* __restrict__ tgt, unsigned* __restrict__ cnt) {
    int i = blockIdx.x * blockDim.x + threadIdx.x;   // over R*E flat
    if (i >= RR * EE) return;
    int r = i / EE;
    atomicAdd(&cnt[(size_t)r * NN + tgt[i]], 1u);    // native global_atomic_add_u32
}

__global__ void rgcn_inv_kernel(const unsigned* __restrict__ cnt, float* __restrict__ invc, int n) {
    int i = blockIdx.x * blockDim.x + threadIdx.x;
    if (i < n) invc[i] = 1.0f / fmaxf((float)cnt[i], 1.0f);
}

// ---------------------------------------------------------------------------
// Exclusive prefix-sum of degrees, one block (256 thr) per relation.
// Chunked Hillis-Steele scan in LDS with a running carry.
// ---------------------------------------------------------------------------
#define SCAN_T 256
__global__ __launch_bounds__(SCAN_T)
void rgcn_scan_kernel(const unsigned* __restrict__ cnt, unsigned* __restrict__ offs) {
    __shared__ unsigned s[SCAN_T];
    const int r = blockIdx.x;
    const unsigned* c = cnt + (size_t)r * NN;
    unsigned* o = offs + (size_t)r * NN;
    unsigned running = 0;
    for (int base = 0; base < NN; base += SCAN_T) {
        const int idx = base + (int)threadIdx.x;
        const unsigned v = (idx < NN) ? c[idx] : 0u;
        s[threadIdx.x] = v;
        __syncthreads();
        #pragma unroll
        for (int d = 1; d < SCAN_T; d <<= 1) {
            const unsigned t = (threadIdx.x >= (unsigned)d) ? s[threadIdx.x - d] : 0u;
            __syncthreads();
            s[threadIdx.x] += t;
            __syncthreads();
        }
        const unsigned incl = s[threadIdx.x];
        if (idx < NN) o[idx] = running + incl - v;   // exclusive
        const unsigned total = s[SCAN_T - 1];
        __syncthreads();
        running += total;
    }
}

// CSR fill: csr[r][offs[t] + cursor[t]++] = src  (u32 cursor atomics, 800K total)
__global__ void rgcn_csrfill_kernel(const int* __restrict__ src, const int* __restrict__ tgt,
                                    const unsigned* __restrict__ offs, unsigned* __restrict__ cursor,
                                    int* __restrict__ csr) {
    int i = blockIdx.x * blockDim.x + threadIdx.x;   // over R*E flat
    if (i >= RR * EE) return;
    const int r = i / EE;
    const int t = tgt[i];
    const unsigned p = atomicAdd(&cursor[(size_t)r * NN + t], 1u);
    csr[(size_t)r * EE + offs[(size_t)r * NN + t] + p] = src[i];
}

// ---------------------------------------------------------------------------
// Aggregation (atomic-free): one wave32 per node. Each lane owns 8 contiguous
// channels (2x float4 -> b128 loads). acc[node] += invc[node] * sum_e Y[src_e].
// ---------------------------------------------------------------------------
__global__ __launch_bounds__(256)
void rgcn_aggregate_kernel(const float* __restrict__ Y, const int* __restrict__ csr,
                           const unsigned* __restrict__ offs, const unsigned* __restrict__ cnt,
                           const float* __restrict__ invc, float* __restrict__ acc) {
    const int lane = threadIdx.x & 31;
    const int node = blockIdx.x * 8 + (threadIdx.x >> 5);
    if (node >= NN) return;
    const unsigned deg = cnt[node];
    if (deg == 0) return;                        // no incoming edges: acc unchanged
    const unsigned beg = offs[node];
    const int cbase = lane * 8;                  // 8 contiguous channels per lane

    float s0 = 0.f, s1 = 0.f, s2 = 0.f, s3 = 0.f;
    float s4 = 0.f, s5 = 0.f, s6 = 0.f, s7 = 0.f;
    for (unsigned e = 0; e < deg; ++e) {
        const int srow = csr[beg + e];           // uniform across wave -> scalar load
        const float4* yp = (const float4*)(Y + (size_t)srow * DH + cbase);
        const float4 y0 = yp[0];                 // b128
        const float4 y1 = yp[1];                 // b128
        s0 += y0.x; s1 += y0.y; s2 += y0.z; s3 += y0.w;
        s4 += y1.x; s5 += y1.y; s6 += y1.z; s7 += y1.w;
    }
    const float w = invc[node];
    float4* ap = (float4*)(acc + (size_t)node * DH + cbase);
    float4 a0 = ap[0];
    float4 a1 = ap[1];
    a0.x += w * s0; a0.y += w * s1; a0.z += w * s2; a0.w += w * s3;
    a1.x += w * s4; a1.y += w * s5; a1.z += w * s6; a1.w += w * s7;
    ap[0] = a0;
    ap[1] = a1;
}

// ---------------------------------------------------------------------------
// Fused ReLU + LayerNorm over 256 channels. One wave32 per node, 8 values/lane,
// mean/var via lane shuffles (wave32!).
// ---------------------------------------------------------------------------
__global__ __launch_bounds__(256)
void rgcn_relu_ln_kernel(const float* __restrict__ acc, const float* __restrict__ g,
                         const float* __restrict__ b, float* __restrict__ out, int n) {
    const int lane = threadIdx.x & 31;
    const int node = blockIdx.x * 8 + (threadIdx.x >> 5);
    if (node >= n) return;
    const float* row = acc + (size_t)node * DH;
    float v[8];
    float sum = 0.0f;
    #pragma unroll
    for (int i = 0; i < 8; ++i) {
        float x = row[i * 32 + lane];
        x = x > 0.0f ? x : 0.0f;                 // ReLU
        v[i] = x;
        sum += x;
    }
    #pragma unroll
    for (int off = 16; off > 0; off >>= 1) sum += __shfl_xor(sum, off, 32);
    const float mu = sum * (1.0f / DH);
    float vs = 0.0f;
    #pragma unroll
    for (int i = 0; i < 8; ++i) { float d = v[i] - mu; vs += d * d; }
    #pragma unroll
    for (int off = 16; off > 0; off >>= 1) vs += __shfl_xor(vs, off, 32);
    const float inv = rsqrtf(vs * (1.0f / DH) + LNE);
    float* orow = out + (size_t)node * DH;
    #pragma unroll
    for (int i = 0; i < 8; ++i) {
        const int c = i * 32 + lane;
        orow[c] = (v[i] - mu) * inv * g[c] + b[c];
    }
}

// ---------------------------------------------------------------------------
extern "C" void kernel_launch(void* const* d_in, const int* in_sizes, int n_in,
                              void* d_out, int out_size, void* d_ws, size_t ws_size,
                              hipStream_t stream) {
    const float* x    = (const float*)d_in[0];
    const float* nw   = (const float*)d_in[1];
    const int*   esrc = (const int*)  d_in[2];   // [R,E]
    const int*   etgt = (const int*)  d_in[3];   // [R,E]
    const float* Wr0  = (const float*)d_in[4];   // [R,768,256]
    const float* Ws0  = (const float*)d_in[5];
    const float* bs0  = (const float*)d_in[6];
    const float* g0   = (const float*)d_in[7];
    const float* b0   = (const float*)d_in[8];
    const float* Wr1  = (const float*)d_in[9];   // [R,256,256]
    const float* Ws1  = (const float*)d_in[10];
    const float* bs1  = (const float*)d_in[11];
    const float* g1   = (const float*)d_in[12];
    const float* b1   = (const float*)d_in[13];
    float* out = (float*)d_out;

    char* ws = (char*)d_ws;
    float*    Y      = (float*)ws;    ws += (size_t)NN * DH * sizeof(float);   // 51.2 MB
    float*    acc    = (float*)ws;    ws += (size_t)NN * DH * sizeof(float);   // 51.2 MB
    float*    h      = (float*)ws;    ws += (size_t)NN * DH * sizeof(float);   // 51.2 MB
    float*    invc   = (float*)ws;    ws += (size_t)RR * NN * sizeof(float);   // 1.6 MB
    unsigned* cntU   = (unsigned*)ws; ws += (size_t)RR * NN * sizeof(unsigned);// 1.6 MB
    unsigned* offs   = (unsigned*)ws; ws += (size_t)RR * NN * sizeof(unsigned);// 1.6 MB
    unsigned* cursor = (unsigned*)ws; ws += (size_t)RR * NN * sizeof(unsigned);// 1.6 MB
    int*      csr    = (int*)ws;                                               // 3.2 MB

    // --- CSR-by-target + 1/max(cnt,1), shared by both layers ---
    {
        const int tot = RR * NN;
        rgcn_fill_u32_kernel<<<(tot + 255) / 256, 256, 0, stream>>>(cntU, 0u, tot);
        rgcn_fill_u32_kernel<<<(tot + 255) / 256, 256, 0, stream>>>(cursor, 0u, tot);
        rgcn_count_kernel<<<(RR * EE + 255) / 256, 256, 0, stream>>>(etgt, cntU);
        rgcn_inv_kernel<<<(tot + 255) / 256, 256, 0, stream>>>(cntU, invc, tot);
        rgcn_scan_kernel<<<RR, SCAN_T, 0, stream>>>(cntU, offs);
        rgcn_csrfill_kernel<<<(RR * EE + 255) / 256, 256, 0, stream>>>(esrc, etgt, offs, cursor, csr);
    }

    const dim3 gblk(256);
    const dim3 ggrid((NN + 127) / 128, DH / 64);
    const int ngrid = (NN + 7) / 8;

    // --- layer 0 ---
    rgcn_gemm_wmma_f32<false, true><<<ggrid, gblk, 0, stream>>>(x, Ws0, bs0, nullptr, acc, NN, DIN);
    for (int r = 0; r < RR; ++r) {
        rgcn_gemm_wmma_f32<true, false><<<ggrid, gblk, 0, stream>>>(
            x, Wr0 + (size_t)r * DIN * DH, nullptr, nw, Y, NN, DIN);
        rgcn_aggregate_kernel<<<ngrid, 256, 0, stream>>>(
            Y, csr + (size_t)r * EE, offs + (size_t)r * NN, cntU + (size_t)r * NN,
            invc + (size_t)r * NN, acc);
    }
    rgcn_relu_ln_kernel<<<ngrid, 256, 0, stream>>>(acc, g0, b0, h, NN);

    // --- layer 1 ---
    rgcn_gemm_wmma_f32<false, true><<<ggrid, gblk, 0, stream>>>(h, Ws1, bs1, nullptr, acc, NN, DH);
    for (int r = 0; r < RR; ++r) {
        rgcn_gemm_wmma_f32<true, false><<<ggrid, gblk, 0, stream>>>(
            h, Wr1 + (size_t)r * DH * DH, nullptr, nw, Y, NN, DH);
        rgcn_aggregate_kernel<<<ngrid, 256, 0, stream>>>(
            Y, csr + (size_t)r * EE, offs + (size_t)r * NN, cntU + (size_t)r * NN,
            invc + (size_t)r * NN, acc);
    }
    rgcn_relu_ln_kernel<<<ngrid, 256, 0, stream>>>(acc, g1, b1, out, NN);
}